// GraphTrans_4879082848570
// MI455X (gfx1250) — compile-verified
//
#include <hip/hip_runtime.h>
#include <math.h>

// ---------------- problem constants (from reference) ----------------
#define HIDDIM 64
#define NHEAD  4
#define DHEAD  16
#define DOUT   32
#define ATT_SCALE 0.25f   // 1/sqrt(16)

typedef __attribute__((ext_vector_type(2))) float v2f;
typedef __attribute__((ext_vector_type(8))) float v8f;

// fp32 WMMA: D(16x16) = A(16x4) * B(4x16) + C   (exact fp32, matches reference)
__device__ __forceinline__ v8f wmma_f32_k4(v2f a, v2f b, v8f c) {
  return __builtin_amdgcn_wmma_f32_16x16x4_f32(
      /*neg_a=*/false, a, /*neg_b=*/false, b,
      /*c_mod=*/(short)0, c, /*reuse_a=*/false, /*reuse_b=*/false);
}

// order-preserving float<->int transform for atomic max on signed ints
__device__ __forceinline__ int f2ord(float f) {
  int i = __float_as_int(f);
  return (i >= 0) ? i : (i ^ 0x7fffffff);
}
__device__ __forceinline__ float ord2f(int i) {
  return __int_as_float((i >= 0) ? i : (i ^ 0x7fffffff));
}
#define NEG_INF_ORD 0x807FFFFF   // f2ord(-inf)

// ---------------- weight transpose: W (64 x ncols) -> Wt (ncols x 64) ----------------
// Makes B-fragment loads contiguous v2f (global_load_b64) instead of 256B-strided b32.
__global__ __launch_bounds__(256) void transpose_w_kernel(
    const float* __restrict__ W, float* __restrict__ Wt, int ncols) {
  const int idx = blockIdx.x * 256 + threadIdx.x;
  const int total = 64 * ncols;
  if (idx >= total) return;
  const int k = idx / ncols;
  const int n = idx - k * ncols;
  Wt[n * 64 + k] = W[idx];
}

// ---------------- scratch init ----------------
__global__ __launch_bounds__(256) void init_scratch_kernel(
    int* __restrict__ mOrd, float* __restrict__ den, float* __restrict__ agg, int n) {
  int idx = blockIdx.x * 256 + threadIdx.x;
  if (idx < n * NHEAD) { mOrd[idx] = NEG_INF_ORD; den[idx] = 0.0f; }
  if (idx < n * HIDDIM) agg[idx] = 0.0f;
}

// ---------------- fused Q/K/V/Skip GEMM ----------------
// Each wave owns one 16-node tile; A-fragments loaded once and reused across all
// four weight matrices (4 matrices x 4 N-tiles = 16 accumulator tiles, 256 WMMAs/wave).
__global__ __launch_bounds__(128) void qkvs_gemm_kernel(
    const float* __restrict__ xin,
    const float* __restrict__ WtQ, const float* __restrict__ WtK,
    const float* __restrict__ WtV, const float* __restrict__ WtS,
    const float* __restrict__ bq, const float* __restrict__ bk,
    const float* __restrict__ bv, const float* __restrict__ bs,
    float* __restrict__ qo, float* __restrict__ ko,
    float* __restrict__ vo, float* __restrict__ so, int ntiles) {
  const int wave = threadIdx.x >> 5;
  const int lane = threadIdx.x & 31;
  const int tile = blockIdx.x * 4 + wave;
  if (tile >= ntiles) return;
  const int node0 = tile * 16;

  const int row = lane & 15;            // A: M index
  const int kb  = (lane >> 4) << 1;     // A/B: K sub-offset 0 or 2
  const int col = lane & 15;            // B/C/D: N index
  const int mb  = (lane >> 4) << 3;     // C/D: M base 0 or 8

  // preload all 16 A-fragments (K = 0..63) for this 16-row tile
  v2f afrag[16];
  const float* ap = xin + (size_t)(node0 + row) * HIDDIM + kb;
#pragma unroll
  for (int kk = 0; kk < 16; ++kk) afrag[kk] = *(const v2f*)(ap + kk * 4);

  const float* Wt[4]   = { WtQ, WtK, WtV, WtS };
  const float* bias[4] = { bq, bk, bv, bs };
  float*       outp[4] = { qo, ko, vo, so };

#pragma unroll
  for (int m = 0; m < 4; ++m) {
    const float* w = Wt[m];
#pragma unroll
    for (int nt = 0; nt < 4; ++nt) {
      const int n0 = nt * 16;
      const float b0 = bias[m][n0 + col];
      v8f acc;
#pragma unroll
      for (int r = 0; r < 8; ++r) acc[r] = b0;     // bias broadcast down rows
      const float* wrow = w + (size_t)(n0 + col) * HIDDIM + kb;  // Wt row = contiguous K
#pragma unroll
      for (int kk = 0; kk < 16; ++kk) {
        const v2f bfrag = *(const v2f*)(wrow + kk * 4);
        acc = wmma_f32_k4(afrag[kk], bfrag, acc);
      }
#pragma unroll
      for (int r = 0; r < 8; ++r)
        outp[m][(size_t)(node0 + mb + r) * HIDDIM + n0 + col] = acc[r];
    }
  }
}

// ---------------- edge pass 1: logits + segment max (atomic) ----------------
__global__ __launch_bounds__(256) void edge_logits_kernel(
    const int* __restrict__ src, const int* __restrict__ dst,
    const float* __restrict__ q, const float* __restrict__ k,
    float* __restrict__ logits, int* __restrict__ mOrd, int e_count) {
  int e = blockIdx.x * 256 + threadIdx.x;
  if (e >= e_count) return;
  const int s = src[e], d = dst[e];
  const float4* qd = (const float4*)(q + (size_t)d * HIDDIM);
  const float4* ks = (const float4*)(k + (size_t)s * HIDDIM);
#pragma unroll
  for (int h = 0; h < NHEAD; ++h) {
    float acc = 0.0f;
#pragma unroll
    for (int i = 0; i < 4; ++i) {
      float4 a = qd[h * 4 + i];
      float4 b = ks[h * 4 + i];
      acc += a.x * b.x + a.y * b.y + a.z * b.z + a.w * b.w;
    }
    const float l = acc * ATT_SCALE;
    logits[(size_t)e * NHEAD + h] = l;
    atomicMax(&mOrd[d * NHEAD + h], f2ord(l));
  }
}

// ---------------- edge pass 2: exp, denominator, weighted aggregate ----------------
__global__ __launch_bounds__(256) void edge_agg_kernel(
    const int* __restrict__ src, const int* __restrict__ dst,
    const float* __restrict__ logits, const int* __restrict__ mOrd,
    const float* __restrict__ v, float* __restrict__ den,
    float* __restrict__ agg, int e_count) {
  int e = blockIdx.x * 256 + threadIdx.x;
  if (e >= e_count) return;
  const int s = src[e], d = dst[e];
  float ex[NHEAD];
#pragma unroll
  for (int h = 0; h < NHEAD; ++h) {
    ex[h] = expf(logits[(size_t)e * NHEAD + h] - ord2f(mOrd[d * NHEAD + h]));
    atomicAdd(&den[d * NHEAD + h], ex[h]);
  }
  const float4* vs = (const float4*)(v + (size_t)s * HIDDIM);
  float* ag = agg + (size_t)d * HIDDIM;
#pragma unroll
  for (int i = 0; i < 16; ++i) {
    const float4 vv = vs[i];
    const float w = ex[i >> 2];
    atomicAdd(ag + i * 4 + 0, w * vv.x);
    atomicAdd(ag + i * 4 + 1, w * vv.y);
    atomicAdd(ag + i * 4 + 2, w * vv.z);
    atomicAdd(ag + i * 4 + 3, w * vv.w);
  }
}

// ---------------- finalize: normalize softmax, +skip, optional ReLU, optional LN ----
// one wave32 per node, 2 channels per lane, shfl_xor reductions for LN stats
__global__ __launch_bounds__(256) void finalize_kernel(
    const float* __restrict__ agg, const float* __restrict__ den,
    const float* __restrict__ skip, const float* __restrict__ ln_g,
    const float* __restrict__ ln_b, float* __restrict__ out,
    int do_relu, int do_ln, int n) {
  const int wave = threadIdx.x >> 5;
  const int lane = threadIdx.x & 31;
  const int node = blockIdx.x * 8 + wave;
  if (node >= n) return;
  const int d0 = lane * 2;
  const int head = d0 >> 4;
  const float dn = den[(size_t)node * NHEAD + head];
  const float inv = (dn > 0.0f) ? (1.0f / dn) : 0.0f;
  const float2 a = *(const float2*)(agg + (size_t)node * HIDDIM + d0);
  const float2 sk = *(const float2*)(skip + (size_t)node * HIDDIM + d0);
  float h0 = a.x * inv + sk.x;
  float h1 = a.y * inv + sk.y;
  if (do_relu) { h0 = fmaxf(h0, 0.0f); h1 = fmaxf(h1, 0.0f); }
  if (do_ln) {
    float ssum = h0 + h1;
#pragma unroll
    for (int off = 16; off > 0; off >>= 1) ssum += __shfl_xor(ssum, off, 32);
    const float mu = ssum * (1.0f / HIDDIM);
    const float dx0 = h0 - mu, dx1 = h1 - mu;
    float vsum = dx0 * dx0 + dx1 * dx1;
#pragma unroll
    for (int off = 16; off > 0; off >>= 1) vsum += __shfl_xor(vsum, off, 32);
    const float r = rsqrtf(vsum * (1.0f / HIDDIM) + 1e-5f);
    h0 = dx0 * r * ln_g[d0] + ln_b[d0];
    h1 = dx1 * r * ln_g[d0 + 1] + ln_b[d0 + 1];
  }
  float2 o; o.x = h0; o.y = h1;
  *(float2*)(out + (size_t)node * HIDDIM + d0) = o;
}

// ---------------- output GEMM: out(N x 32) = h(N x 64) @ Wo + bo (WMMA) ----------------
__global__ __launch_bounds__(256) void out_gemm_kernel(
    const float* __restrict__ h, const float* __restrict__ WtO,
    const float* __restrict__ bo, float* __restrict__ out, int ntiles) {
  const int wave = threadIdx.x >> 5;
  const int lane = threadIdx.x & 31;
  const int tile = blockIdx.x * 8 + wave;
  if (tile >= ntiles) return;
  const int node0 = tile * 16;
  const int row = lane & 15;
  const int kb  = (lane >> 4) << 1;
  const int col = lane & 15;
  const int mb  = (lane >> 4) << 3;

  v2f afrag[16];
  const float* ap = h + (size_t)(node0 + row) * HIDDIM + kb;
#pragma unroll
  for (int kk = 0; kk < 16; ++kk) afrag[kk] = *(const v2f*)(ap + kk * 4);

#pragma unroll
  for (int nt = 0; nt < 2; ++nt) {
    const int n0 = nt * 16;
    const float b0 = bo[n0 + col];
    v8f acc;
#pragma unroll
    for (int r = 0; r < 8; ++r) acc[r] = b0;
    const float* wrow = WtO + (size_t)(n0 + col) * HIDDIM + kb;
#pragma unroll
    for (int kk = 0; kk < 16; ++kk) {
      const v2f bfrag = *(const v2f*)(wrow + kk * 4);
      acc = wmma_f32_k4(afrag[kk], bfrag, acc);
    }
#pragma unroll
    for (int r = 0; r < 8; ++r)
      out[(size_t)(node0 + mb + r) * DOUT + n0 + col] = acc[r];
  }
}

// ---------------- host orchestration ----------------
extern "C" void kernel_launch(void* const* d_in, const int* in_sizes, int n_in,
                              void* d_out, int out_size, void* d_ws, size_t ws_size,
                              hipStream_t stream) {
  const int N = in_sizes[0] / HIDDIM;   // x is (N, 64)
  const int E = in_sizes[1] / 2;        // edge_index is (2, E)

  const float* x  = (const float*)d_in[0];
  const int*   ei = (const int*)d_in[1];
  const int* src = ei;
  const int* dst = ei + E;

  const float *Wq[3], *Wk[3], *Wv[3], *Ws[3], *bq[3], *bk[3], *bv[3], *bs[3];
  for (int L = 0; L < 3; ++L) {
    const int base = 3 + L * 8;
    Wq[L] = (const float*)d_in[base + 0];
    Wk[L] = (const float*)d_in[base + 1];
    Wv[L] = (const float*)d_in[base + 2];
    Ws[L] = (const float*)d_in[base + 3];
    bq[L] = (const float*)d_in[base + 4];
    bk[L] = (const float*)d_in[base + 5];
    bv[L] = (const float*)d_in[base + 6];
    bs[L] = (const float*)d_in[base + 7];
  }
  const float* ln_g[2] = { (const float*)d_in[27], (const float*)d_in[29] };
  const float* ln_b[2] = { (const float*)d_in[28], (const float*)d_in[30] };
  const float* Wo = (const float*)d_in[31];
  const float* bo = (const float*)d_in[32];

  // workspace layout (256B-aligned slabs)
  const size_t nf = (size_t)N * HIDDIM;
  char* p = (char*)d_ws;
  auto carve = [&](size_t bytes) {
    char* r = p;
    p += (bytes + 255) & ~(size_t)255;
    return r;
  };
  float* q     = (float*)carve(nf * sizeof(float));
  float* k     = (float*)carve(nf * sizeof(float));
  float* v     = (float*)carve(nf * sizeof(float));
  float* skip  = (float*)carve(nf * sizeof(float));
  float* agg   = (float*)carve(nf * sizeof(float));
  float* t     = (float*)carve(nf * sizeof(float));
  float* logit = (float*)carve((size_t)E * NHEAD * sizeof(float));
  float* den   = (float*)carve((size_t)N * NHEAD * sizeof(float));
  int*   mOrd  = (int*)  carve((size_t)N * NHEAD * sizeof(int));
  // transposed weights: 12 x (64x64) + Wo (32x64)
  float* WtQ[3], *WtK[3], *WtV[3], *WtS[3];
  for (int L = 0; L < 3; ++L) {
    WtQ[L] = (float*)carve(64 * 64 * sizeof(float));
    WtK[L] = (float*)carve(64 * 64 * sizeof(float));
    WtV[L] = (float*)carve(64 * 64 * sizeof(float));
    WtS[L] = (float*)carve(64 * 64 * sizeof(float));
  }
  float* WtO = (float*)carve((size_t)DOUT * 64 * sizeof(float));

  // ---- transpose all weight matrices once (tiny) ----
  const int tb64 = (64 * 64 + 255) / 256;
  for (int L = 0; L < 3; ++L) {
    transpose_w_kernel<<<tb64, 256, 0, stream>>>(Wq[L], WtQ[L], 64);
    transpose_w_kernel<<<tb64, 256, 0, stream>>>(Wk[L], WtK[L], 64);
    transpose_w_kernel<<<tb64, 256, 0, stream>>>(Wv[L], WtV[L], 64);
    transpose_w_kernel<<<tb64, 256, 0, stream>>>(Ws[L], WtS[L], 64);
  }
  transpose_w_kernel<<<(64 * DOUT + 255) / 256, 256, 0, stream>>>(Wo, WtO, DOUT);

  const int ntiles      = N / 16;                    // N = 100000, multiple of 16
  const int init_blocks = (int)((nf + 255) / 256);
  const int edge_blocks = (E + 255) / 256;
  const int node_blocks = (N + 7) / 8;

  const float* xin = x;
  for (int L = 0; L < 3; ++L) {
    init_scratch_kernel<<<init_blocks, 256, 0, stream>>>(mOrd, den, agg, N);
    qkvs_gemm_kernel<<<(ntiles + 3) / 4, 128, 0, stream>>>(
        xin, WtQ[L], WtK[L], WtV[L], WtS[L], bq[L], bk[L], bv[L], bs[L],
        q, k, v, skip, ntiles);
    edge_logits_kernel<<<edge_blocks, 256, 0, stream>>>(src, dst, q, k, logit, mOrd, E);
    edge_agg_kernel<<<edge_blocks, 256, 0, stream>>>(src, dst, logit, mOrd, v, den, agg, E);
    const int do_relu = (L >= 1) ? 1 : 0;       // layers 1,2 have ReLU
    const int do_ln   = (L <= 1) ? 1 : 0;       // ln1 after layer0, ln2 after layer1
    const float* g = do_ln ? ln_g[L] : (const float*)nullptr;
    const float* b = do_ln ? ln_b[L] : (const float*)nullptr;
    finalize_kernel<<<node_blocks, 256, 0, stream>>>(agg, den, skip, g, b, t,
                                                     do_relu, do_ln, N);
    xin = t;
  }
  out_gemm_kernel<<<(ntiles + 7) / 8, 256, 0, stream>>>(t, WtO, bo, (float*)d_out, ntiles);
}